// glimpse_3d_2920577761577
// MI455X (gfx1250) — compile-verified
//
#include <hip/hip_runtime.h>
#include <hip/hip_bf16.h>

typedef float v2f __attribute__((ext_vector_type(2)));
typedef float v8f __attribute__((ext_vector_type(8)));

#define BATCH 16
#define HDIM 96
#define GPATCH 38
#define SP0 (38*38*38)        /* 54872 */
#define EPSBN 1e-5f

static inline int cdiv(long a, int b) { return (int)((a + b - 1) / b); }

// ---------------------------------------------------------------------------
// 1) per-image fill value: mean of x[b, 94:96, 94:96, 94:96] (C==1)
// ---------------------------------------------------------------------------
__global__ void fill_k(const float* __restrict__ x, float* __restrict__ fill) {
    int b = threadIdx.x;
    if (b >= BATCH) return;
    const float* xb = x + (size_t)b * HDIM * HDIM * HDIM;
    float s = 0.f;
    for (int i = 94; i < 96; ++i)
        for (int j = 94; j < 96; ++j)
            for (int k = 94; k < 96; ++k)
                s += xb[(i * HDIM + j) * HDIM + k];
    fill[b] = s * (1.0f / 8.0f);
}

// ---------------------------------------------------------------------------
// 2) foveation: phi (B, 2, 38,38,38). scale0 = direct 38^3 patch,
//    scale1 = 2x2x2 avg-pool of the 76^3 patch. OOB -> fill[b].
// ---------------------------------------------------------------------------
__global__ void foveate_k(const float* __restrict__ x, const float* __restrict__ l,
                          const float* __restrict__ fill, float* __restrict__ phi) {
    long t = (long)blockIdx.x * blockDim.x + threadIdx.x;
    const long total = (long)BATCH * 2 * SP0;
    if (t >= total) return;
    int p  = (int)(t % SP0);
    int s  = (int)((t / SP0) % 2);
    int b  = (int)(t / (2 * SP0));
    int d  = p % GPATCH;
    int w  = (p / GPATCH) % GPATCH;
    int h  = p / (GPATCH * GPATCH);

    float l0 = l[b * 3 + 0], l1 = l[b * 3 + 1], l2 = l[b * 3 + 2];
    int dn0 = (int)floorf(0.5f * (l0 + 1.0f) * (float)HDIM);
    int dn1 = (int)floorf(0.5f * (l1 + 1.0f) * (float)HDIM);
    int dn2 = (int)floorf(0.5f * (l2 + 1.0f) * (float)HDIM);

    const float* xb = x + (size_t)b * HDIM * HDIM * HDIM;
    float fv = fill[b];
    float val;
    if (s == 0) {
        int st0 = dn0 - 19, st1 = dn1 - 19, st2 = dn2 - 19;
        int hh = st0 + h, ww = st1 + w, dd = st2 + d;
        bool inb = (hh >= 0) & (hh < HDIM) & (ww >= 0) & (ww < HDIM) &
                   (dd >= 0) & (dd < HDIM);
        val = inb ? xb[((long)hh * HDIM + ww) * HDIM + dd] : fv;
    } else {
        int st0 = dn0 - 38, st1 = dn1 - 38, st2 = dn2 - 38;
        float acc = 0.f;
        #pragma unroll
        for (int dh = 0; dh < 2; ++dh)
        #pragma unroll
        for (int dw = 0; dw < 2; ++dw)
        #pragma unroll
        for (int dd2 = 0; dd2 < 2; ++dd2) {
            int hh = st0 + 2 * h + dh, ww = st1 + 2 * w + dw, dd = st2 + 2 * d + dd2;
            bool inb = (hh >= 0) & (hh < HDIM) & (ww >= 0) & (ww < HDIM) &
                       (dd >= 0) & (dd < HDIM);
            acc += inb ? xb[((long)hh * HDIM + ww) * HDIM + dd] : fv;
        }
        val = acc * 0.125f;
    }
    phi[t] = val;
}

// ---------------------------------------------------------------------------
// 3) generic VALID 3x3x3 conv, NCDHW in, OIDHW weights.
// ---------------------------------------------------------------------------
__global__ void conv3d_k(const float* __restrict__ in, const float* __restrict__ wgt,
                         const float* __restrict__ bias, float* __restrict__ out,
                         int Cin, int Cout, int din, int dout, long in_bstride) {
    long t = (long)blockIdx.x * blockDim.x + threadIdx.x;
    int osp = dout * dout * dout;
    long total = (long)BATCH * Cout * osp;
    if (t >= total) return;
    int j  = (int)(t % osp);
    int co = (int)((t / osp) % Cout);
    int b  = (int)(t / ((long)osp * Cout));
    int ox = j % dout;
    int oy = (j / dout) % dout;
    int oz = j / (dout * dout);

    int isp = din * din * din;
    const float* ib = in + (long)b * in_bstride;
    float acc = bias[co];
    for (int ci = 0; ci < Cin; ++ci) {
        const float* ic = ib + (long)ci * isp;
        const float* wc = wgt + ((long)co * Cin + ci) * 27;
        #pragma unroll
        for (int kz = 0; kz < 3; ++kz)
        #pragma unroll
        for (int ky = 0; ky < 3; ++ky)
        #pragma unroll
        for (int kx = 0; kx < 3; ++kx)
            acc += ic[((long)(oz + kz) * din + (oy + ky)) * din + (ox + kx)] *
                   wc[(kz * 3 + ky) * 3 + kx];
    }
    out[t] = acc;
}

// ---------------------------------------------------------------------------
// 4) per-channel batch stats (biased variance), one block per channel
// ---------------------------------------------------------------------------
__global__ void bn_stats_k(const float* __restrict__ in, float* __restrict__ mean,
                           float* __restrict__ var, int C, int sp) {
    int c = blockIdx.x;
    long n = (long)BATCH * sp;
    float s = 0.f, sq = 0.f;
    for (long i = threadIdx.x; i < n; i += blockDim.x) {
        int b = (int)(i / sp);
        int j = (int)(i % sp);
        float v = in[((long)b * C + c) * sp + j];
        s += v; sq += v * v;
    }
    __shared__ float ss[256], ssq[256];
    ss[threadIdx.x] = s; ssq[threadIdx.x] = sq;
    __syncthreads();
    for (int st = 128; st > 0; st >>= 1) {
        if (threadIdx.x < st) { ss[threadIdx.x] += ss[threadIdx.x + st];
                                ssq[threadIdx.x] += ssq[threadIdx.x + st]; }
        __syncthreads();
    }
    if (threadIdx.x == 0) {
        float m = ss[0] / (float)n;
        mean[c] = m;
        var[c]  = ssq[0] / (float)n - m * m;
    }
}

// ---------------------------------------------------------------------------
// 5) fused BN + ReLU + 2x2x2 maxpool
// ---------------------------------------------------------------------------
__global__ void bn_relu_pool_k(const float* __restrict__ in, const float* __restrict__ mean,
                               const float* __restrict__ var, const float* __restrict__ g,
                               const float* __restrict__ bb, float* __restrict__ out,
                               int C, int din) {
    int dout = din / 2;
    int osp = dout * dout * dout, isp = din * din * din;
    long t = (long)blockIdx.x * blockDim.x + threadIdx.x;
    long total = (long)BATCH * C * osp;
    if (t >= total) return;
    int j = (int)(t % osp);
    int c = (int)((t / osp) % C);
    int b = (int)(t / ((long)osp * C));
    int ox = j % dout, oy = (j / dout) % dout, oz = j / (dout * dout);

    float m = mean[c];
    float sc = g[c] * rsqrtf(var[c] + EPSBN);
    float sh = bb[c];
    const float* ib = in + ((long)b * C + c) * isp;
    float best = 0.f;   // post-ReLU values are >= 0
    #pragma unroll
    for (int dz = 0; dz < 2; ++dz)
    #pragma unroll
    for (int dy = 0; dy < 2; ++dy)
    #pragma unroll
    for (int dx = 0; dx < 2; ++dx) {
        float v = ib[((long)(2 * oz + dz) * din + (2 * oy + dy)) * din + (2 * ox + dx)];
        v = fmaxf((v - m) * sc + sh, 0.f);
        best = fmaxf(best, v);
    }
    out[t] = best;
}

// ---------------------------------------------------------------------------
// 6) BN + ReLU in place (layer 3)
// ---------------------------------------------------------------------------
__global__ void bn_relu_k(float* __restrict__ a, const float* __restrict__ mean,
                          const float* __restrict__ var, const float* __restrict__ g,
                          const float* __restrict__ bb, int C, int sp) {
    long t = (long)blockIdx.x * blockDim.x + threadIdx.x;
    long total = (long)BATCH * C * sp;
    if (t >= total) return;
    int c = (int)((t / sp) % C);
    float v = a[t];
    a[t] = fmaxf((v - mean[c]) * g[c] * rsqrtf(var[c] + EPSBN) + bb[c], 0.f);
}

// ---------------------------------------------------------------------------
// 7) BN (no relu) + flatten into feats[b][col_off + c*sp + j] (layer 4)
// ---------------------------------------------------------------------------
__global__ void bn_flatten_k(const float* __restrict__ in, const float* __restrict__ mean,
                             const float* __restrict__ var, const float* __restrict__ g,
                             const float* __restrict__ bb, float* __restrict__ feats,
                             int C, int sp, int col_off) {
    long t = (long)blockIdx.x * blockDim.x + threadIdx.x;
    long total = (long)BATCH * C * sp;
    if (t >= total) return;
    int j = (int)(t % sp);
    int c = (int)((t / sp) % C);
    int b = (int)(t / ((long)sp * C));
    float v = in[t];
    v = (v - mean[c]) * g[c] * rsqrtf(var[c] + EPSBN) + bb[c];
    feats[(long)b * 8192 + col_off + c * sp + j] = v;
}

// ---------------------------------------------------------------------------
// 8) l_out = relu(l @ fc2_w^T + fc2_b)   (16 x 1024, K=3)
// ---------------------------------------------------------------------------
__global__ void fc2_k(const float* __restrict__ l, const float* __restrict__ w,
                      const float* __restrict__ bias, float* __restrict__ lout) {
    int t = blockIdx.x * blockDim.x + threadIdx.x;
    if (t >= BATCH * 1024) return;
    int n = t % 1024, b = t / 1024;
    float acc = bias[n];
    acc += l[b * 3 + 0] * w[n * 3 + 0];
    acc += l[b * 3 + 1] * w[n * 3 + 1];
    acc += l[b * 3 + 2] * w[n * 3 + 2];
    lout[t] = fmaxf(acc, 0.f);
}

// ---------------------------------------------------------------------------
// 9) Final FC on the matrix pipe (exact f32 via V_WMMA_F32_16X16X4_F32).
//    One block (4 waves) per 16-col N tile; each wave reduces a K/4 slice
//    with TWO independent accumulator chains (breaks WMMA D->C serialization),
//    partials combined through LDS.
//    A fragment (16x4): lane<16 -> M=lane, K={0,1}; lane>=16 -> M=lane-16,
//    K={2,3}. B fragment (4x16) mirrors with N=lane%16. C/D v8f: row
//    m = reg + 8*(lane>=16), col n = lane%16.
// ---------------------------------------------------------------------------
__global__ __launch_bounds__(128)
void fc_wmma_k(const float* __restrict__ feats, const float* __restrict__ Wg,
               const float* __restrict__ bg, const float* __restrict__ lout,
               float* __restrict__ out) {
    __shared__ float red[4][256];          // 4 waves x (8 regs * 32 lanes)
    const int tid  = threadIdx.x;
    const int wave = tid >> 5;             // 0..3   K-slice
    const int lane = tid & 31;
    const int ntile = blockIdx.x;          // 0..63
    const int half = lane >> 4;            // 0 / 1
    const int l16  = lane & 15;
    const int koff = half * 2;             // K sub-offset for this half-wave
    const int kbase = wave * 2048;         // this wave's K slice

    const float* __restrict__ arow = feats + (long)l16 * 8192 + kbase + koff;
    const float* __restrict__ brow =
        Wg + ((long)(ntile * 16 + l16)) * 8192 + kbase + koff;

    v8f acc0 = {}, acc1 = {};
    #pragma unroll 4
    for (int k = 0; k < 2048; k += 8) {
        if ((k & 511) == 0 && k + 1024 < 2048)
            __builtin_prefetch(brow + k + 1024, 0, 0);   // global_prefetch_b8
        v2f a0 = *(const v2f*)(arow + k);
        v2f b0 = *(const v2f*)(brow + k);
        v2f a1 = *(const v2f*)(arow + k + 4);
        v2f b1 = *(const v2f*)(brow + k + 4);
        acc0 = __builtin_amdgcn_wmma_f32_16x16x4_f32(
            false, a0, false, b0, (short)0, acc0, false, false);
        acc1 = __builtin_amdgcn_wmma_f32_16x16x4_f32(
            false, a1, false, b1, (short)0, acc1, false, false);
    }

    #pragma unroll
    for (int r = 0; r < 8; ++r)
        red[wave][r * 32 + lane] = acc0[r] + acc1[r];
    __syncthreads();

    if (wave == 0) {
        const int col = ntile * 16 + l16;
        const float bias = bg[col];
        #pragma unroll
        for (int r = 0; r < 8; ++r) {
            float v = red[0][r * 32 + lane] + red[1][r * 32 + lane] +
                      red[2][r * 32 + lane] + red[3][r * 32 + lane];
            int m = half * 8 + r;
            float what = fmaxf(v + bias, 0.f);
            out[m * 1024 + col] = fmaxf(what * lout[m * 1024 + col], 0.f);
        }
    }
}

// ---------------------------------------------------------------------------
extern "C" void kernel_launch(void* const* d_in, const int* in_sizes, int n_in,
                              void* d_out, int out_size, void* d_ws, size_t ws_size,
                              hipStream_t stream) {
    const float* x     = (const float*)d_in[0];
    const float* l     = (const float*)d_in[1];
    const float* c1_w  = (const float*)d_in[2];
    const float* c1_b  = (const float*)d_in[3];
    const float* bn1_g = (const float*)d_in[4];
    const float* bn1_b = (const float*)d_in[5];
    const float* c2_w  = (const float*)d_in[6];
    const float* c2_b  = (const float*)d_in[7];
    const float* bn2_g = (const float*)d_in[8];
    const float* bn2_b = (const float*)d_in[9];
    const float* c3_w  = (const float*)d_in[10];
    const float* c3_b  = (const float*)d_in[11];
    const float* bn3_g = (const float*)d_in[12];
    const float* bn3_b = (const float*)d_in[13];
    const float* c4_w  = (const float*)d_in[14];
    const float* c4_b  = (const float*)d_in[15];
    const float* bn4_g = (const float*)d_in[16];
    const float* bn4_b = (const float*)d_in[17];
    const float* fcg_w = (const float*)d_in[18];
    const float* fcg_b = (const float*)d_in[19];
    const float* fc2_w = (const float*)d_in[20];
    const float* fc2_b = (const float*)d_in[21];

    float* ws = (float*)d_ws;
    size_t off = 0;
    float* fill  = ws + off; off += 16;
    float* phi   = ws + off; off += (size_t)BATCH * 2 * SP0;          // 1.76M
    float* c1o   = ws + off; off += (size_t)BATCH * 8 * 46656;        // 5.97M
    float* p1o   = ws + off; off += (size_t)BATCH * 8 * 5832;
    float* c2o   = ws + off; off += (size_t)BATCH * 16 * 4096;
    float* p2o   = ws + off; off += (size_t)BATCH * 16 * 512;
    float* c3o   = ws + off; off += (size_t)BATCH * 32 * 216;
    float* c4o   = ws + off; off += (size_t)BATCH * 64 * 64;
    float* mean  = ws + off; off += 64;
    float* var   = ws + off; off += 64;
    float* lout  = ws + off; off += (size_t)BATCH * 1024;
    float* feats = ws + off; off += (size_t)BATCH * 8192;

    (void)in_sizes; (void)n_in; (void)out_size; (void)ws_size;

    fill_k<<<1, 32, 0, stream>>>(x, fill);
    foveate_k<<<cdiv((long)BATCH * 2 * SP0, 256), 256, 0, stream>>>(x, l, fill, phi);

    for (int s = 0; s < 2; ++s) {
        // layer 1: 1->8, 38->36, pool -> 18
        conv3d_k<<<cdiv((long)BATCH * 8 * 46656, 256), 256, 0, stream>>>(
            phi + (size_t)s * SP0, c1_w, c1_b, c1o, 1, 8, 38, 36, (long)2 * SP0);
        bn_stats_k<<<8, 256, 0, stream>>>(c1o, mean, var, 8, 46656);
        bn_relu_pool_k<<<cdiv((long)BATCH * 8 * 5832, 256), 256, 0, stream>>>(
            c1o, mean, var, bn1_g, bn1_b, p1o, 8, 36);

        // layer 2: 8->16, 18->16, pool -> 8
        conv3d_k<<<cdiv((long)BATCH * 16 * 4096, 256), 256, 0, stream>>>(
            p1o, c2_w, c2_b, c2o, 8, 16, 18, 16, (long)8 * 5832);
        bn_stats_k<<<16, 256, 0, stream>>>(c2o, mean, var, 16, 4096);
        bn_relu_pool_k<<<cdiv((long)BATCH * 16 * 512, 256), 256, 0, stream>>>(
            c2o, mean, var, bn2_g, bn2_b, p2o, 16, 16);

        // layer 3: 16->32, 8->6
        conv3d_k<<<cdiv((long)BATCH * 32 * 216, 256), 256, 0, stream>>>(
            p2o, c3_w, c3_b, c3o, 16, 32, 8, 6, (long)16 * 512);
        bn_stats_k<<<32, 256, 0, stream>>>(c3o, mean, var, 32, 216);
        bn_relu_k<<<cdiv((long)BATCH * 32 * 216, 256), 256, 0, stream>>>(
            c3o, mean, var, bn3_g, bn3_b, 32, 216);

        // layer 4: 32->64, 6->4, BN only, flatten
        conv3d_k<<<cdiv((long)BATCH * 64 * 64, 256), 256, 0, stream>>>(
            c3o, c4_w, c4_b, c4o, 32, 64, 6, 4, (long)32 * 216);
        bn_stats_k<<<64, 256, 0, stream>>>(c4o, mean, var, 64, 64);
        bn_flatten_k<<<cdiv((long)BATCH * 64 * 64, 256), 256, 0, stream>>>(
            c4o, mean, var, bn4_g, bn4_b, feats, 64, 64, s * 4096);
    }

    fc2_k<<<cdiv(BATCH * 1024, 256), 256, 0, stream>>>(l, fc2_w, fc2_b, lout);
    fc_wmma_k<<<64, 128, 0, stream>>>(feats, fcg_w, fcg_b, lout, (float*)d_out);
}